// FrameReducer_16020228014889
// MI455X (gfx1250) — compile-verified
//
#include <hip/hip_runtime.h>
#include <math.h>
#include <stdint.h>

// Problem constants (fixed by the harness reference)
#define N_  16
#define T_  2048
#define C_  512
#define V_  1024
#define SLICES 8          // rank kernel: blocks per batch row
#define TPB_A  256        // T_/SLICES must equal TPB_A

// ---------------------------------------------------------------------------
// Kernel A: per-frame rank of blank log-prob + keep-mask.
// grid = (N, SLICES), block = 256. Each block stages the full 2048-entry
// blank-lp column in LDS (all lanes read the same LDS word per iteration ->
// conflict-free broadcast), each thread ranks exactly one t.
// ---------------------------------------------------------------------------
__global__ void fr_rank_mask(const float* __restrict__ ctc,
                             const int*  __restrict__ x_lens,
                             const int*  __restrict__ y_lens,
                             const int*  __restrict__ blank_id_p,
                             unsigned char* __restrict__ mask_ws)
{
    __shared__ float lp[T_];
    const int n   = blockIdx.x;
    const int bid = blank_id_p[0];

    for (int i = threadIdx.x; i < T_; i += TPB_A)
        lp[i] = ctc[((size_t)n * T_ + i) * V_ + bid];
    __syncthreads();

    const int   t = blockIdx.y * TPB_A + threadIdx.x;
    const float v = lp[t];

    // rank[t] = #{lp[u] > v} + #{u < t : lp[u] == v}   (stable desc sort rank)
    int cnt = 0;
#pragma unroll 8
    for (int u = 0; u < T_; ++u) {
        const float w = lp[u];
        cnt += ((w > v) || ((w == v) && (u < t))) ? 1 : 0;
    }

    const int   x_len = x_lens[n];
    const int   limit = T_ - y_lens[n];                    // limit_lens
    const float LOG_TH = -0.05129329438755058f;            // ln(0.95)

    const bool non_blank = (v < LOG_TH) && (t < x_len);    // & ~padding
    const bool keep      = non_blank || (cnt >= limit);    // | ~limit_mask
    mask_ws[n * T_ + t] = keep ? 1u : 0u;
}

// ---------------------------------------------------------------------------
// Kernel B: per-row stable compaction indices. grid = N, block = 256,
// each thread owns 8 consecutive t. Writes order[n][pos] = t for kept frames
// and out_lens (int for the gather kernel, float into the output tail).
// ---------------------------------------------------------------------------
__global__ void fr_scan(const unsigned char* __restrict__ mask_ws,
                        int*   __restrict__ order_ws,
                        int*   __restrict__ outlen_ws,
                        float* __restrict__ out_tail)
{
    __shared__ int csum[256];
    __shared__ int coff[256];
    const int n = blockIdx.x;
    const unsigned char* m = mask_ws + n * T_;

    const int base = threadIdx.x * 8;
    unsigned char loc[8];
    int s = 0;
#pragma unroll
    for (int k = 0; k < 8; ++k) { loc[k] = m[base + k]; s += loc[k]; }
    csum[threadIdx.x] = s;
    __syncthreads();

    if (threadIdx.x == 0) {
        int acc = 0;
        for (int i = 0; i < 256; ++i) { coff[i] = acc; acc += csum[i]; }
        outlen_ws[n] = acc;
        out_tail[n]  = (float)acc;     // out_lens in the harness output dtype
    }
    __syncthreads();

    int pos = coff[threadIdx.x];
#pragma unroll
    for (int k = 0; k < 8; ++k)
        if (loc[k]) order_ws[n * T_ + (pos++)] = base + k;
}

// ---------------------------------------------------------------------------
// Kernel C: bulk gather, one 2KB output row per 128-thread block.
// Kept rows ride the CDNA5 async copy engine: per lane
//   global_load_async_to_lds_b128 (16B)  ->  s_wait_asynccnt 0
//   global_store_async_from_lds_b128 (16B)
// Branch is block-uniform so EXEC is all-ones for the async ops (ISA req).
// Dropped rows are zero-filled with vector stores (d_out is poisoned).
// Async stores are drained by the implicit wait-idle at s_endpgm.
// ---------------------------------------------------------------------------
__global__ void fr_gather(const float* __restrict__ x,
                          const int*   __restrict__ order_ws,
                          const int*   __restrict__ outlen_ws,
                          float*       __restrict__ out)
{
    __shared__ __align__(16) float tile[C_];     // 128 lanes * 16B = 2KB
    const int r   = blockIdx.x;
    const int n   = r / T_;
    const int j   = r % T_;
    const int tid = threadIdx.x;                 // 0..127

    const int len = outlen_ws[n];
    float* dst = out + ((size_t)n * T_ + j) * C_ + tid * 4;

    if (j < len) {
        const int src = order_ws[n * T_ + j];
        const float* g = x + ((size_t)n * T_ + src) * C_ + tid * 4;

        // AS(3) pointer value == workgroup-relative LDS byte offset; the flat
        // pointer's low 32 bits carry it through (LDS aperture in the high bits).
        const uint32_t lds   = (uint32_t)(size_t)&tile[tid * 4];
        const unsigned long long ga = (unsigned long long)(size_t)g;
        const unsigned long long da = (unsigned long long)(size_t)dst;

        asm volatile("global_load_async_to_lds_b128 %0, %1, off"
                     :: "v"(lds), "v"(ga) : "memory");
        asm volatile("s_wait_asynccnt 0" ::: "memory");
        asm volatile("global_store_async_from_lds_b128 %0, %1, off"
                     :: "v"(da), "v"(lds) : "memory");
    } else {
        *(float4*)dst = make_float4(0.f, 0.f, 0.f, 0.f);
    }
}

// ---------------------------------------------------------------------------
extern "C" void kernel_launch(void* const* d_in, const int* in_sizes, int n_in,
                              void* d_out, int out_size, void* d_ws, size_t ws_size,
                              hipStream_t stream)
{
    const float* x        = (const float*)d_in[0];   // (N,T,C) f32
    const int*   x_lens   = (const int*)  d_in[1];   // (N,)
    const float* ctc      = (const float*)d_in[2];   // (N,T,V) f32
    const int*   y_lens   = (const int*)  d_in[3];   // (N,)
    const int*   blank_id = (const int*)  d_in[4];   // scalar
    float*       out      = (float*)d_out;           // (N,T,C) flat + N lens

    // Workspace layout: [mask: N*T bytes][order: N*T ints][outlen: N ints]
    unsigned char* mask_ws   = (unsigned char*)d_ws;
    int*           order_ws  = (int*)((char*)d_ws + ((N_ * T_ + 255) & ~255));
    int*           outlen_ws = order_ws + (size_t)N_ * T_;

    dim3 gridA(N_, SLICES);
    fr_rank_mask<<<gridA, TPB_A, 0, stream>>>(ctc, x_lens, y_lens, blank_id, mask_ws);
    fr_scan<<<N_, 256, 0, stream>>>(mask_ws, order_ws, outlen_ws,
                                    out + (size_t)N_ * T_ * C_);
    fr_gather<<<N_ * T_, 128, 0, stream>>>(x, order_ws, outlen_ws, out);
}